// DeepGMM_21311627723347
// MI455X (gfx1250) — compile-verified
//
#include <hip/hip_runtime.h>

typedef __attribute__((ext_vector_type(2))) float v2f;
typedef __attribute__((ext_vector_type(8))) float v8f;
typedef __attribute__((ext_vector_type(4))) unsigned u32x4;
typedef __attribute__((ext_vector_type(4))) int i32x4;
typedef __attribute__((ext_vector_type(8))) int i32x8;

constexpr int NB = 4096;   // batch
constexpr int KC = 32;     // clusters
constexpr int DC = 32;     // dims
constexpr int SC = 8;      // samples
constexpr float LOG2PI = 1.8378770664093453f;

// ---- workspace layout (float offsets) ----
constexpr size_t OFF_SENC   = 0;        // [32,32] phi_enc_sigma
constexpr size_t OFF_INVENC = 1024;     // [32,32]
constexpr size_t OFF_CHENC  = 2048;     // [32,32] cholesky(enc_sigma)
constexpr size_t OFF_CHY    = 3072;     // [32,32] cholesky(dec_sigma)
constexpr size_t OFF_DSIG   = 4096;     // [32,32] dec_sigma
constexpr size_t OFF_SCAL   = 5120;     // [0]=hld_enc [1]=hld_y [2]=Tsum
constexpr size_t OFF_CZ     = 5152;     // [K] c_z constants
constexpr size_t OFF_BZ     = 5184;     // [K,D] A_z mu_k
constexpr size_t OFF_M2     = 6208;     // [K,D] sigma_tilde t2
constexpr size_t OFF_ST     = 7232;     // [K,32,32] sigma_tilde
constexpr size_t OFF_AZ     = 40000;    // [K,32,32] inv(enc_sigma+phi_sigma)
constexpr size_t OFF_TSIG   = 72768;    // [K,32,32] theta_sigmas
constexpr size_t OFF_PSIG   = 105536;   // [K,32,32] phi_sigmas
constexpr size_t OFF_SCR1   = 138304;   // [K,32,32] scratch
constexpr size_t OFF_SCR2   = 171072;   // [K,32,32] scratch
constexpr size_t OFF_EMU    = 203840;   // [N,D] phi_enc_mu
constexpr size_t OFF_T1     = 334912;   // [N,D] t1
constexpr size_t OFF_ZL     = 465984;   // [N,K] z_logits
constexpr size_t OFF_LSE    = 597056;   // [N]
constexpr size_t OFF_Z      = 601152;   // [N*S,K] gumbel-softmax z
constexpr size_t OFF_DZL    = 1649728;  // [N*S] dot(z_logits, z)
constexpr size_t OFF_DTP    = 1682496;  // [N*S] dot(theta_pis, z)
constexpr size_t OFF_AZT    = 1715264;  // [32,1024] A_z transposed: [e][k*32+d]

// ================= TDM (Tensor Data Mover) 1D contiguous load ===============
#if defined(__has_builtin)
#  if __has_builtin(__builtin_amdgcn_tensor_load_to_lds)
#    define HAVE_TDM 1
#  else
#    define HAVE_TDM 0
#  endif
#else
#  define HAVE_TDM 0
#endif

#if HAVE_TDM
// D# per CDNA5 ISA: 1D tile of nelem f32 elements.
__device__ __forceinline__ void tdm_load_lin(void* lds_dst, const void* gsrc,
                                             unsigned nelem) {
  unsigned long long ga = (unsigned long long)(size_t)gsrc;
  unsigned lo = (unsigned)(size_t)lds_dst;   // LDS aperture low bits == LDS byte offset
  u32x4 g0 = { 1u,                                   // count=1, is_restore=0, gather=0
               lo,                                   // lds_addr
               (unsigned)ga,                         // global_addr[31:0]
               (unsigned)((ga >> 32) & 0x1FFFFFFull) | (2u << 30) }; // addr[56:32] | type=2
  i32x8 g1 = { (int)0x00020000u,                     // mask=0, data_size=2 (4 bytes)
               (int)((nelem & 0xFFFFu) << 16),       // tensor_dim0[15:0]
               (int)(((nelem >> 16) & 0xFFFFu) | (1u << 16)), // dim0[31:16] | tensor_dim1=1
               (int)((nelem & 0xFFFFu) << 16),       // tile_dim0 = nelem
               0,                                    // tile_dim1=0 (1D), tile_dim2=0
               (int)nelem,                           // tensor_dim0_stride lo
               0, 0 };
  i32x4 gz = { 0, 0, 0, 0 };
#if defined(__clang_major__) && (__clang_major__ >= 23)
  i32x8 gz8 = { 0, 0, 0, 0, 0, 0, 0, 0 };
  __builtin_amdgcn_tensor_load_to_lds(g0, g1, gz, gz, gz8, 0);
#else
  __builtin_amdgcn_tensor_load_to_lds(g0, g1, gz, gz, 0);
#endif
}
#endif

// ================= serial dense helpers (lane-private, global mem) =========
__device__ void chol32(float* A) {           // in-place lower cholesky
  for (int j = 0; j < DC; j++) {
    float s = A[j*DC+j];
    for (int t = 0; t < j; t++) { float l = A[j*DC+t]; s -= l*l; }
    float Ljj = sqrtf(fmaxf(s, 1e-30f));
    A[j*DC+j] = Ljj;
    float r = 1.0f/Ljj;
    for (int i = j+1; i < DC; i++) {
      float v = A[i*DC+j];
      for (int t = 0; t < j; t++) v -= A[i*DC+t]*A[j*DC+t];
      A[i*DC+j] = v*r;
    }
  }
}
__device__ void trinv32(const float* L, float* W) {  // W = L^{-1}, zero upper
  for (int j = 0; j < DC; j++) {
    W[j*DC+j] = 1.0f/L[j*DC+j];
    for (int i = j+1; i < DC; i++) {
      float s = 0.f;
      for (int t = j; t < i; t++) s -= L[i*DC+t]*W[t*DC+j];
      W[i*DC+j] = s / L[i*DC+i];
    }
    for (int i = 0; i < j; i++) W[i*DC+j] = 0.f;
  }
}
__device__ void wtw32(const float* W, float* A) {    // A = W^T W (SPD inverse)
  for (int d = 0; d < DC; d++)
    for (int e = 0; e < DC; e++) {
      int lo = d > e ? d : e;
      float s = 0.f;
      for (int i = lo; i < DC; i++) s += W[i*DC+d]*W[i*DC+e];
      A[d*DC+e] = s;
    }
}

// ================= K1: batch-independent precompute (1 wave) ================
__global__ void k1_precompute(
    const float* __restrict__ encA, const float* __restrict__ decA,
    const float* __restrict__ phi_mus, const float* __restrict__ phi_diags,
    const float* __restrict__ phi_factors, const float* __restrict__ phi_pis,
    const float* __restrict__ th_diags, const float* __restrict__ th_factors,
    float* __restrict__ ws)
{
  const int l = threadIdx.x;               // lane = cluster k
  float* Senc = ws + OFF_SENC;
  float* Dsig = ws + OFF_DSIG;
  for (int e = 0; e < DC; e++) {           // row l of both shared covariances
    float s = 0.f, sd = 0.f;
    for (int t = 0; t < DC; t++) { s += encA[l*DC+t]*encA[e*DC+t]; sd += decA[l*DC+t]*decA[e*DC+t]; }
    Senc[l*DC+e] = s*(1.0f/DC) + (l==e ? 0.1f : 0.f);
    Dsig[l*DC+e] = sd*(1.0f/DC) + (l==e ? 0.1f : 0.f);
  }
  float* psig = ws + OFF_PSIG + (size_t)l*1024;
  float* tsig = ws + OFF_TSIG + (size_t)l*1024;
  for (int d = 0; d < DC; d++)
    for (int e = 0; e < DC; e++) {
      float vp = phi_factors[l*DC+d]*phi_factors[l*DC+e];
      float vt = th_factors[l*DC+d]*th_factors[l*DC+e];
      if (d == e) { vp += phi_diags[l*DC+d]+1e-6f; vt += th_diags[l*DC+d]+1e-6f; }
      psig[d*DC+e] = vp; tsig[d*DC+e] = vt;
    }
  __syncthreads();
  // chol_z[k] = chol(enc_sigma + phi_sigma_k)  (scratch1)
  float* zk = ws + OFF_SCR1 + (size_t)l*1024;
  for (int i = 0; i < 1024; i++) zk[i] = Senc[i] + psig[i];
  chol32(zk);
  float hldz = 0.f;
  for (int j = 0; j < DC; j++) hldz += logf(zk[j*DC+j]);
  if (l == 0) {
    float* ce = ws + OFF_CHENC;
    for (int i = 0; i < 1024; i++) ce[i] = Senc[i];
    chol32(ce);
    float h = 0.f; for (int j = 0; j < DC; j++) h += logf(ce[j*DC+j]);
    ws[OFF_SCAL+0] = h;
  }
  if (l == 1) {
    float* cy = ws + OFF_CHY;
    for (int i = 0; i < 1024; i++) cy[i] = Dsig[i];
    chol32(cy);
    float h = 0.f; for (int j = 0; j < DC; j++) h += logf(cy[j*DC+j]);
    ws[OFF_SCAL+1] = h;
  }
  if (l == 2) ws[OFF_SCAL+2] = 0.f;        // zero global term accumulator
  // A_z = (enc_sigma + phi_sigma)^{-1} via triangular inverse
  float* w2 = ws + OFF_SCR2 + (size_t)l*1024;
  trinv32(zk, w2);
  float* Az = ws + OFF_AZ + (size_t)l*1024;
  wtw32(w2, Az);
  // transposed copy for the WMMA quadratic-form GEMM (B[e][k*32+d])
  for (int d = 0; d < DC; d++)
    for (int e = 0; e < DC; e++)
      ws[OFF_AZT + (size_t)e*1024 + l*DC + d] = Az[d*DC+e];
  float dot = 0.f;
  for (int d = 0; d < DC; d++) {
    float s = 0.f;
    for (int e = 0; e < DC; e++) s += Az[d*DC+e]*phi_mus[l*DC+e];
    ws[OFF_BZ + l*DC + d] = s;
    dot += s*phi_mus[l*DC+d];
  }
  ws[OFF_CZ+l] = logf(phi_pis[l]) - 0.5f*DC*LOG2PI - hldz - 0.5f*dot;
  __syncthreads();
  if (l == 0) {                            // inv_enc from chol_enc
    trinv32(ws + OFF_CHENC, zk);
    wtw32(zk, ws + OFF_INVENC);
  }
  __syncthreads();
  // sigma_tilde = inv(inv_enc + inv_phi_k); inv_phi via Sherman-Morrison
  const float* inve = ws + OFF_INVENC;
  float invd[DC], fk[DC];
  float a = 0.f;
  for (int d = 0; d < DC; d++) {
    invd[d] = 1.0f/(phi_diags[l*DC+d]+1e-6f);
    fk[d]   = phi_factors[l*DC+d];
    a += fk[d]*fk[d]*invd[d];
  }
  float alpha = 1.0f/(1.0f+a);
  for (int d = 0; d < DC; d++)
    for (int e = 0; e < DC; e++) {
      float ip = ((d==e) ? invd[d] : 0.f) - alpha*invd[d]*fk[d]*fk[e]*invd[e];
      zk[d*DC+e] = inve[d*DC+e] + ip;
    }
  chol32(zk);
  trinv32(zk, w2);
  float* St = ws + OFF_ST + (size_t)l*1024;
  wtw32(w2, St);
  // t2 = inv_phi mu_k (Sherman-Morrison), m2 = sigma_tilde t2
  float fm = 0.f;
  for (int e = 0; e < DC; e++) fm += fk[e]*invd[e]*phi_mus[l*DC+e];
  float t2v[DC];
  for (int d = 0; d < DC; d++) t2v[d] = invd[d]*phi_mus[l*DC+d] - alpha*invd[d]*fk[d]*fm;
  for (int d = 0; d < DC; d++) {
    float s = 0.f;
    for (int e = 0; e < DC; e++) s += St[d*DC+e]*t2v[e];
    ws[OFF_M2 + l*DC + d] = s;
  }
}

// ================= WMMA f32 16x16x4 helpers ================================
__device__ __forceinline__ v8f wmma_f32(v2f a, v2f b, v8f c) {
  // D = A(16x4 f32) * B(4x16 f32) + C(16x16 f32)
  return __builtin_amdgcn_wmma_f32_16x16x4_f32(false, a, false, b, (short)0, c, false, false);
}
// A frag: lanes0-15 row=lane, K = 4*kk+{0,1}; lanes16-31 row=lane-16, K = 4*kk+{2,3}
__device__ __forceinline__ v2f frag_a(const float* src, int ld, int kk, int lane) {
  int r = lane & 15, o = (lane >> 4) * 2;
  v2f a; a.x = src[r*ld + kk*4 + o]; a.y = src[r*ld + kk*4 + o + 1];
  return a;
}
// B frag: lanes0-15 col=col0+lane, K = 4*kk+{0,1}; lanes16-31 K = 4*kk+{2,3}
__device__ __forceinline__ v2f frag_b(const float* src, int ld, int kk, int col0, int lane) {
  int c = col0 + (lane & 15), o = (lane >> 4) * 2;
  v2f b; b.x = src[(kk*4+o)*ld + c]; b.y = src[(kk*4+o+1)*ld + c];
  return b;
}
// D store: vgpr r -> row (lane<16 ? r : r+8), col = col0 + lane%16
__device__ __forceinline__ void store_d16(float* dst, int ld, int col0, int lane, v8f c) {
  int cc = col0 + (lane & 15), rb = (lane >> 4) * 8;
#pragma unroll
  for (int r = 0; r < 8; r++) dst[(rb+r)*ld + cc] = c[r];
}

// ================= K2: encoder GEMMs (WMMA) ================================
__global__ void k2_encoder(const float* __restrict__ Y, const float* __restrict__ encW,
                           const float* __restrict__ encb, float* __restrict__ ws) {
  __shared__ float Et[16*DC];
  const int lane = threadIdx.x;
  const int n0 = blockIdx.x * 16;
  const float* Ys = Y + (size_t)n0*DC;
  float* emu = ws + OFF_EMU + (size_t)n0*DC;
#pragma unroll
  for (int ct = 0; ct < 2; ct++) {               // E = Y@W + b
    float bv = encb[ct*16 + (lane & 15)];
    v8f c = {bv,bv,bv,bv,bv,bv,bv,bv};
#pragma unroll
    for (int kk = 0; kk < 8; kk++)
      c = wmma_f32(frag_a(Ys, DC, kk, lane), frag_b(encW, DC, kk, ct*16, lane), c);
    store_d16(emu, DC, ct*16, lane, c);
    store_d16(Et,  DC, ct*16, lane, c);
  }
  __syncthreads();
  const float* inve = ws + OFF_INVENC;           // t1 = E @ inv_enc (symmetric)
  float* t1 = ws + OFF_T1 + (size_t)n0*DC;
#pragma unroll
  for (int ct = 0; ct < 2; ct++) {
    v8f c = {0,0,0,0,0,0,0,0};
#pragma unroll
    for (int kk = 0; kk < 8; kk++)
      c = wmma_f32(frag_a(Et, DC, kk, lane), frag_b(inve, DC, kk, ct*16, lane), c);
    store_d16(t1, DC, ct*16, lane, c);
  }
}

// ====== K3a: z_logits via WMMA quadratic form P = mu-tile @ A_z^T ==========
__global__ void k3a_logits(float* __restrict__ ws) {
  __shared__ float mu16[16][DC];
  __shared__ float P[16][DC*33];      // padded (k -> pitch 33)
  const int t = threadIdx.x, lane = t & 31, wave = t >> 5;
  const int n0 = blockIdx.x * 16;
#if HAVE_TDM
  if (wave == 0) {
    tdm_load_lin(&mu16[0][0], ws + OFF_EMU + (size_t)n0*DC, 16*DC);
    __builtin_amdgcn_s_wait_tensorcnt(0);
  }
#else
  for (int i = t; i < 16*DC; i += 128)
    mu16[i >> 5][i & 31] = ws[OFF_EMU + (size_t)n0*DC + i];
#endif
  __syncthreads();
  v2f aM[8];
#pragma unroll
  for (int kk = 0; kk < 8; kk++) aM[kk] = frag_a(&mu16[0][0], DC, kk, lane);
  const float* B = ws + OFF_AZT;      // [32 e][1024 = k*32+d]
  for (int c0 = 0; c0 < 16; c0 += 2) {  // two col-tiles -> independent chains
    int colA = (wave*16 + c0) * 16, colB = colA + 16;
    v8f a0 = {0,0,0,0,0,0,0,0}, a1 = {0,0,0,0,0,0,0,0};
#pragma unroll
    for (int kk = 0; kk < 8; kk++) {
      a0 = wmma_f32(aM[kk], frag_b(B, 1024, kk, colA, lane), a0);
      a1 = wmma_f32(aM[kk], frag_b(B, 1024, kk, colB, lane), a1);
    }
    int rb = (lane >> 4) * 8;
    int cA = colA + (lane & 15), cB = colB + (lane & 15);
#pragma unroll
    for (int r = 0; r < 8; r++) {
      P[rb+r][(cA >> 5)*33 + (cA & 31)] = a0[r];
      P[rb+r][(cB >> 5)*33 + (cB & 31)] = a1[r];
    }
  }
  __syncthreads();
  // z_logits[n,k] = c_k + sum_d mu_d * (b_z[k,d] - 0.5*P[n,k,d])
  for (int i = t; i < 512; i += 128) {
    int r = i >> 5, k = i & 31;
    float s = 0.f;
    for (int d = 0; d < DC; d++)
      s += mu16[r][d] * (ws[OFF_BZ + k*DC + d] - 0.5f * P[r][k*33 + d]);
    ws[OFF_ZL + (size_t)(n0 + r)*KC + k] = ws[OFF_CZ + k] + s;
  }
}

// ====== K3b: lse, gumbel-softmax z, small dots (wave per n) ================
__global__ void k3b_gumbel(const float* __restrict__ u_gum, const float* __restrict__ th_pis,
                           const float* __restrict__ temp, float* __restrict__ ws) {
  const int n = blockIdx.x;
  const int k = threadIdx.x;                     // lane = cluster
  float zl = ws[OFF_ZL + (size_t)n*KC + k];
  float mx = zl;
  for (int o = 16; o > 0; o >>= 1) mx = fmaxf(mx, __shfl_xor(mx, o, 32));
  float ex = expf(zl - mx), sm = ex;
  for (int o = 16; o > 0; o >>= 1) sm += __shfl_xor(sm, o, 32);
  if (k == 0) ws[OFF_LSE+n] = mx + logf(sm);
  const float T = temp[0], tp = th_pis[k];
  for (int s = 0; s < SC; s++) {
    float u = u_gum[((size_t)s*NB + n)*KC + k];
    u = fminf(fmaxf(u, 1e-6f), 1.0f - 1e-6f);
    float g = -logf(-logf(u));
    float w = (zl + g) / T;
    float wm = w;
    for (int o = 16; o > 0; o >>= 1) wm = fmaxf(wm, __shfl_xor(wm, o, 32));
    float e2 = expf(w - wm), ss = e2;
    for (int o = 16; o > 0; o >>= 1) ss += __shfl_xor(ss, o, 32);
    float zv = e2 / ss;
    int ns = n*SC + s;
    ws[OFF_Z + (size_t)ns*KC + k] = zv;
    float d1 = zl*zv, d2 = tp*zv;
    for (int o = 16; o > 0; o >>= 1) { d1 += __shfl_xor(d1, o, 32); d2 += __shfl_xor(d2, o, 32); }
    if (k == 0) { ws[OFF_DZL+ns] = d1; ws[OFF_DTP+ns] = d2; }
  }
}

// shuffle-broadcast forward substitution; returns ||L^{-1} d||^2 (same on all lanes)
__device__ __forceinline__ float mahal(const float* L, int ld, float di, int lane) {
  float acc = 0.f, M = 0.f;
  for (int j = 0; j < DC; j++) {
    float tmp = di - acc;
    float yj = __shfl(tmp, j, 32) * (1.0f/L[j*ld+j]);
    M += yj*yj;
    if (lane > j) acc += L[lane*ld+j]*yj;
  }
  return M;
}

// ================= K4: fused mixture GEMMs + cholesky + log-probs ===========
__global__ void k4_fused(const float* __restrict__ Y, const float* __restrict__ eps,
                         const float* __restrict__ decW, const float* __restrict__ decb,
                         const float* __restrict__ th_mus, const float* __restrict__ phi_mus,
                         float* __restrict__ ws) {
  __shared__ float zt[16][DC];        // z rows (GEMM A)
  __shared__ float t1r[16][DC];
  __shared__ float epr[16][DC];
  __shared__ float xb[16][DC];        // sampled x
  __shared__ float mut[16][DC];       // mixed mean (per type)
  __shared__ float mix[16][DC*33];    // mixed covariance, pitch 33 (bank-pad)
  __shared__ float lp_y[16], lp_q[16], lp_t[16], lp_p[16], sterm[16];
  const int t = threadIdx.x, lane = t & 31, wave = t >> 5;
  const int ns0 = blockIdx.x * 16;
#if HAVE_TDM
  if (wave == 0) {                    // async DMA of contiguous staging tiles
    tdm_load_lin(&zt[0][0],  ws + OFF_Z + (size_t)ns0*KC, 16*KC);
    tdm_load_lin(&epr[0][0], eps + (size_t)ns0*DC,        16*DC);
  }
#else
  for (int i = t; i < 16*DC; i += 128) {
    int r = i >> 5, d = i & 31;
    zt[r][d]  = ws[OFF_Z + (size_t)(ns0 + r)*KC + d];
    epr[r][d] = eps[(size_t)(ns0 + r)*DC + d];
  }
#endif
  for (int i = t; i < 16*DC; i += 128) {
    int r = i >> 5, d = i & 31;
    int n = (ns0 + r) >> 3;           // S == 8
    t1r[r][d] = ws[OFF_T1 + (size_t)n*DC + d];
  }
  // warm caches for the later covariance panels
  __builtin_prefetch(ws + OFF_TSIG + (size_t)t*256, 0, 1);
  __builtin_prefetch(ws + OFF_PSIG + (size_t)t*256, 0, 1);
#if HAVE_TDM
  if (wave == 0) __builtin_amdgcn_s_wait_tensorcnt(0);
#endif
  __syncthreads();
  v2f aZ[8];
#pragma unroll
  for (int kk = 0; kk < 8; kk++) aZ[kk] = frag_a(&zt[0][0], DC, kk, lane);

  for (int ty = 0; ty < 3; ty++) {    // 0: sigma_tilde, 1: theta, 2: phi
    const float* B = (ty == 0) ? ws+OFF_ST : (ty == 1) ? ws+OFF_TSIG : ws+OFF_PSIG;
    __syncthreads();
    // mixed[16,1024] = z[16,32] x sigflat[32,1024]; wave handles 16 col-tiles
    for (int c0 = 0; c0 < 16; c0 += 2) {
      int colA = (wave*16 + c0) * 16, colB = colA + 16;
      v8f a0 = {0,0,0,0,0,0,0,0}, a1 = {0,0,0,0,0,0,0,0};
#pragma unroll
      for (int kk = 0; kk < 8; kk++) {
        a0 = wmma_f32(aZ[kk], frag_b(B, 1024, kk, colA, lane), a0);
        a1 = wmma_f32(aZ[kk], frag_b(B, 1024, kk, colB, lane), a1);
      }
      int rb = (lane >> 4) * 8;
      int cA = colA + (lane & 15), cB = colB + (lane & 15);
#pragma unroll
      for (int r = 0; r < 8; r++) {
        mix[rb+r][(cA >> 5)*33 + (cA & 31)] = a0[r];
        mix[rb+r][(cB >> 5)*33 + (cB & 31)] = a1[r];
      }
    }
    __syncthreads();
    // mixed means
    if (ty == 0) {                    // _mu_tilde = Sig~ @ t1[n] + z@m2
      for (int i = t; i < 512; i += 128) {
        int r = i >> 5, d = i & 31;
        float s = 0.f;
        for (int e = 0; e < DC; e++) s += mix[r][d*33+e]*t1r[r][e];
        for (int k2 = 0; k2 < KC; k2++) s += zt[r][k2]*ws[OFF_M2 + k2*DC + d];
        mut[r][d] = s;
      }
    } else {
      const float* mus = (ty == 1) ? th_mus : phi_mus;
      for (int i = t; i < 512; i += 128) {
        int r = i >> 5, d = i & 31;
        float s = 0.f;
        for (int k2 = 0; k2 < KC; k2++) s += zt[r][k2]*mus[k2*DC + d];
        mut[r][d] = s;
      }
    }
    __syncthreads();
    // in-place cholesky of 16 matrices; wave owns 4, lane = row
    for (int j = 0; j < DC; j++) {
      for (int mi = 0; mi < 4; mi++) {
        int m = wave*4 + mi, i2 = lane;
        float ajj = mix[m][j*33+j];
        float Ljj = sqrtf(fmaxf(ajj, 1e-30f));
        float rj = 1.0f/Ljj, rj2 = rj*rj;
        float aij = (i2 > j) ? mix[m][i2*33+j] : 0.f;
        if (i2 > j) {                 // trailing update from OLD column values
          for (int tt = j+1; tt <= i2; tt++) {
            float atj = mix[m][tt*33+j];
            mix[m][i2*33+tt] -= aij*atj*rj2;
          }
          mix[m][i2*33+j] = aij*rj;   // then write scaled column
        }
        if (i2 == j) mix[m][j*33+j] = Ljj;
      }
      __syncthreads();
    }
    if (ty == 0) {                    // x = mu~ + Lx eps ; logq_x ; logp_y
      for (int mi = 0; mi < 4; mi++) {
        int m = wave*4 + mi, d = lane;
        float s = mut[m][d];
        for (int j = 0; j <= d; j++) s += mix[m][d*33+j]*epr[m][j];
        xb[m][d] = s;
      }
      __syncthreads();
      for (int mi = 0; mi < 4; mi++) {
        int m = wave*4 + mi, d = lane, n = (ns0 + m) >> 3;
        float dq2 = xb[m][d] - ws[OFF_EMU + (size_t)n*DC + d];
        float Mq = mahal(ws+OFF_CHENC, DC, dq2, lane);
        float s = decb[d];
        for (int e = 0; e < DC; e++) s += xb[m][e]*decW[e*DC + d];
        float dy = Y[(size_t)n*DC + d] - s;
        float My = mahal(ws+OFF_CHY, DC, dy, lane);
        if (lane == 0) {
          lp_q[m] = -0.5f*(DC*LOG2PI + Mq) - ws[OFF_SCAL+0];
          lp_y[m] = -0.5f*(DC*LOG2PI + My) - ws[OFF_SCAL+1];
        }
      }
    } else {                          // logp under mixed theta / phi
      for (int mi = 0; mi < 4; mi++) {
        int m = wave*4 + mi, d = lane;
        float lg = logf(mix[m][d*33+d]);
        for (int o = 16; o > 0; o >>= 1) lg += __shfl_xor(lg, o, 32);
        float dd = xb[m][d] - mut[m][d];
        float M = mahal(&mix[m][0], 33, dd, lane);
        if (lane == 0) {
          float lp = -0.5f*(DC*LOG2PI + M) - lg;
          if (ty == 1) lp_t[m] = lp; else lp_p[m] = lp;
        }
      }
    }
    __syncthreads();
  }
  if (t < 16) {
    int ns = ns0 + t;
    sterm[t] = lp_y[t] - lp_q[t] + lp_t[t] + ws[OFF_DTP+ns] - lp_p[t] - ws[OFF_DZL+ns];
  }
  __syncthreads();
  if (t == 0) {
    float s = 0.f;
    for (int i = 0; i < 16; i++) s += sterm[i];
    atomicAdd(ws + OFF_SCAL + 2, s);
  }
}

// ================= K5: loss[n] = -(Tsum/S + lse[n]) ========================
__global__ void k5_finalize(float* __restrict__ out, const float* __restrict__ ws) {
  int n = blockIdx.x*blockDim.x + threadIdx.x;
  if (n < NB) out[n] = -(ws[OFF_SCAL+2]*(1.0f/SC) + ws[OFF_LSE+n]);
}

extern "C" void kernel_launch(void* const* d_in, const int* in_sizes, int n_in,
                              void* d_out, int out_size, void* d_ws, size_t ws_size,
                              hipStream_t stream) {
  (void)in_sizes; (void)n_in; (void)out_size; (void)ws_size;
  const float* Y           = (const float*)d_in[0];
  const float* u_gum       = (const float*)d_in[1];
  const float* eps         = (const float*)d_in[2];
  const float* encW        = (const float*)d_in[3];
  const float* encb        = (const float*)d_in[4];
  const float* encA        = (const float*)d_in[5];
  const float* decW        = (const float*)d_in[6];
  const float* decb        = (const float*)d_in[7];
  const float* decA        = (const float*)d_in[8];
  const float* phi_mus     = (const float*)d_in[9];
  const float* phi_diags   = (const float*)d_in[10];
  const float* phi_factors = (const float*)d_in[11];
  const float* phi_pis     = (const float*)d_in[12];
  const float* th_mus      = (const float*)d_in[13];
  const float* th_diags    = (const float*)d_in[14];
  const float* th_factors  = (const float*)d_in[15];
  const float* th_pis      = (const float*)d_in[16];
  const float* temp        = (const float*)d_in[17];
  float* out = (float*)d_out;
  float* ws  = (float*)d_ws;

  hipLaunchKernelGGL(k1_precompute, dim3(1), dim3(32), 0, stream,
                     encA, decA, phi_mus, phi_diags, phi_factors, phi_pis,
                     th_diags, th_factors, ws);
  hipLaunchKernelGGL(k2_encoder, dim3(NB/16), dim3(32), 0, stream, Y, encW, encb, ws);
  hipLaunchKernelGGL(k3a_logits, dim3(NB/16), dim3(128), 0, stream, ws);
  hipLaunchKernelGGL(k3b_gumbel, dim3(NB), dim3(32), 0, stream, u_gum, th_pis, temp, ws);
  hipLaunchKernelGGL(k4_fused, dim3(NB*SC/16), dim3(128), 0, stream,
                     Y, eps, decW, decb, th_mus, phi_mus, ws);
  hipLaunchKernelGGL(k5_finalize, dim3((NB+255)/256), dim3(256), 0, stream, out, ws);
}